// CrossNetwork_31765578121355
// MI455X (gfx1250) — compile-verified
//
#include <hip/hip_runtime.h>
#include <hip/hip_bf16.h>

typedef float v2f __attribute__((ext_vector_type(2)));
typedef float v8f __attribute__((ext_vector_type(8)));

namespace {
constexpr int kD        = 2048;   // feature dim
constexpr int kL        = 6;      // layers
constexpr int kTileRows = 16;     // WMMA M
constexpr int kThreads  = 256;    // 8 waves (wave32)
constexpr int kWaves    = kThreads / 32;
constexpr int kChunk    = kD / kWaves;   // K per wave = 256
}

// ---------------------------------------------------------------------------
// Pre-kernel: ws[0..5] = dcum[l] = (sum_{i<l} b_i) . W_l ; ws[6..7]=0 ;
//             ws[8..8+2047] = bsum[j] = sum_i b[i][j]
// ---------------------------------------------------------------------------
__global__ __launch_bounds__(kThreads)
void cross_pre_kernel(const float* __restrict__ W,
                      const float* __restrict__ b,
                      float* __restrict__ ws) {
    __shared__ float red[kThreads];
    const int t  = threadIdx.x;
    const int c0 = t * (kD / kThreads);          // 8 columns per thread
    float pref[kD / kThreads];
    #pragma unroll
    for (int j = 0; j < kD / kThreads; ++j) pref[j] = 0.0f;

    for (int l = 0; l < kL; ++l) {
        float p = 0.0f;
        #pragma unroll
        for (int j = 0; j < kD / kThreads; ++j)
            p += pref[j] * W[l * kD + c0 + j];
        __syncthreads();            // protect red reuse from previous iter
        red[t] = p;
        __syncthreads();
        for (int s = kThreads / 2; s > 0; s >>= 1) {
            if (t < s) red[t] += red[t + s];
            __syncthreads();
        }
        if (t == 0) ws[l] = red[0]; // dcum[0] == 0 naturally (pref starts 0)
        #pragma unroll
        for (int j = 0; j < kD / kThreads; ++j)
            pref[j] += b[l * kD + c0 + j];
    }
    if (t < 2) ws[6 + t] = 0.0f;
    #pragma unroll
    for (int j = 0; j < kD / kThreads; ++j)
        ws[8 + c0 + j] = pref[j];
}

// ---------------------------------------------------------------------------
// Main kernel: one 16-row tile per block.
//   Phase 1: C = X_tile @ W^T (padded to N=16) via V_WMMA_F32_16X16X4_F32,
//            K split across 8 waves, manually unrolled x4 so 8 global_load_b64
//            batch ahead of 4 back-to-back WMMAs (max MLP per wave).
//   Phase 2: per-row scalar recurrence a_{l+1} = a_l + (a_l*c_l + dcum_l).
//   Phase 3: out = a * x + bsum, coalesced float4 stream (x re-read from L2).
// ---------------------------------------------------------------------------
__global__ __launch_bounds__(kThreads)
void cross_wmma_kernel(const float* __restrict__ x,
                       const float* __restrict__ W,
                       const float* __restrict__ ws,
                       float* __restrict__ out) {
    __shared__ float cred[kWaves * 8 * 32];   // per-wave partial C tiles
    __shared__ float csc[kTileRows * kL];     // reduced C[M][N], N<6
    __shared__ float arow[kTileRows];         // final per-row scalar

    const int t    = threadIdx.x;
    const int lane = t & 31;
    const int wave = t >> 5;
    const int r0   = blockIdx.x * kTileRows;

    // A-matrix (16x4 f32) lane mapping: lanes 0-15 -> rows, K = k..k+1;
    // lanes 16-31 -> same rows, K = k+2..k+3.  B (4x16) mirrors on N.
    const int   mrow   = lane & 15;           // row index for A / N index for B
    const int   koff   = (lane >> 4) << 1;    // 0 or 2
    const bool  nvalid = mrow < kL;           // only 6 of 16 N columns are real
    const float wscale = nvalid ? 1.0f : 0.0f;
    const int   wr     = nvalid ? mrow : 0;   // clamp: branch-free valid address

    const v2f* xv2 = (const v2f*)(x + (size_t)(r0 + mrow) * kD);
    const v2f* wv2 = (const v2f*)(W + (size_t)wr * kD);

    v8f c = {};
    const int j0 = (wave * kChunk + koff) >> 1;      // v2f index (step 2 = 4 K)
    #pragma unroll 1
    for (int j = j0; j < j0 + kChunk / 2; j += 8) {  // 16 K per outer iter
        v2f a[4], bm[4];
        #pragma unroll
        for (int u = 0; u < 4; ++u) {                // 8 batched b64 loads
            a[u]  = xv2[j + 2 * u];
            bm[u] = wv2[j + 2 * u];
        }
        #pragma unroll
        for (int u = 0; u < 4; ++u) {
            bm[u].x *= wscale;                       // v_pk_mul_f32
            bm[u].y *= wscale;
            // D = A(16x4,f32) * B(4x16,f32) + C ; exact fp32 math
            c = __builtin_amdgcn_wmma_f32_16x16x4_f32(false, a[u], false, bm[u],
                                                      (short)0, c, false, false);
        }
    }
    #pragma unroll
    for (int v = 0; v < 8; ++v)
        cred[(wave * 8 + v) * 32 + lane] = c[v];
    __syncthreads();

    // Reduce partial C tiles over waves; C/D layout: VGPR v holds M=v
    // (lanes 0-15) and M=v+8 (lanes 16-31), N = lane&15.
    if (t < kTileRows * kL) {
        const int M  = t / kL;
        const int N  = t % kL;
        const int v  = M & 7;
        const int ln = ((M >> 3) << 4) | N;
        float s = 0.0f;
        #pragma unroll
        for (int w = 0; w < kWaves; ++w)
            s += cred[(w * 8 + v) * 32 + ln];
        csc[t] = s;
    }
    __syncthreads();

    // Per-row scalar recurrence (16 threads).
    if (t < kTileRows) {
        float a = 1.0f;
        #pragma unroll
        for (int l = 0; l < kL; ++l)
            a += a * csc[t * kL + l] + ws[l];   // s_l = a*c_l + dcum_l
        arow[t] = a;
    }
    __syncthreads();

    // Streaming output: out = a * x + bsum  (x re-read is L2-resident).
    const float4* bsum4 = (const float4*)(ws + 8);
    const float4* xg    = (const float4*)(x   + (size_t)r0 * kD);
    float4*       og    = (float4*)      (out + (size_t)r0 * kD);
    #pragma unroll 4
    for (int i = 0; i < (kTileRows * kD / 4) / kThreads; ++i) {
        const int idx = t + i * kThreads;       // 0..8191 (row*512 + c4)
        const int row = idx >> 9;
        const int c4  = idx & 511;
        const float av = arow[row];
        const float4 xv = xg[idx];
        const float4 bs = bsum4[c4];
        float4 o;
        o.x = fmaf(av, xv.x, bs.x);
        o.y = fmaf(av, xv.y, bs.y);
        o.z = fmaf(av, xv.z, bs.z);
        o.w = fmaf(av, xv.w, bs.w);
        og[idx] = o;
    }
}

// ---------------------------------------------------------------------------
extern "C" void kernel_launch(void* const* d_in, const int* in_sizes, int n_in,
                              void* d_out, int out_size, void* d_ws, size_t ws_size,
                              hipStream_t stream) {
    const float* x = (const float*)d_in[0];   // [B, D]
    const float* W = (const float*)d_in[1];   // [L, D]
    const float* b = (const float*)d_in[2];   // [L, D]
    float* out = (float*)d_out;               // [B, D]
    float* ws  = (float*)d_ws;                // needs (8 + D) floats

    const int B = in_sizes[0] / kD;           // 16384

    cross_pre_kernel<<<1, kThreads, 0, stream>>>(W, b, ws);
    cross_wmma_kernel<<<B / kTileRows, kThreads, 0, stream>>>(x, W, ws, out);
}